// ResidualVectorQuantizer_25615184953911
// MI455X (gfx1250) — compile-verified
//
#include <hip/hip_runtime.h>

#define NN 8192
#define DD 256
#define LL 4
#define KK 1024
#define EE 8
#define MAX_TILES (NN / 16 + EE)   // 520: upper bound on padded 16-row tiles

typedef __attribute__((ext_vector_type(2))) float v2f;
typedef __attribute__((ext_vector_type(8))) float v8f;

__device__ __forceinline__ unsigned int fkey32(float f) {
  // monotonic float -> uint mapping (handles negatives)
  unsigned int b = __float_as_uint(f);
  return (b & 0x80000000u) ? ~b : (b | 0x80000000u);
}

__device__ __forceinline__ unsigned long long shfl_xor_u64(unsigned long long v, int m) {
  unsigned int lo = (unsigned int)(v & 0xffffffffull);
  unsigned int hi = (unsigned int)(v >> 32);
  lo = __shfl_xor(lo, m, 32);
  hi = __shfl_xor(hi, m, 32);
  return (((unsigned long long)hi) << 32) | (unsigned long long)lo;
}

// ---------------------------------------------------------------- init
__global__ void k_init(float* __restrict__ xq, int* __restrict__ perm,
                       int* __restrict__ meta) {
  int i = blockIdx.x * blockDim.x + threadIdx.x;
  if (i < NN * DD) xq[i] = 0.0f;
  if (i < NN + 256) perm[i] = -1;
  if (i < 32) meta[i] = 0;
}

// ---------------------------------------------------------------- ||c||^2
__global__ void k_c2(const float* __restrict__ cb, float* __restrict__ c2) {
  int wid = blockIdx.x * 8 + (threadIdx.x >> 5);   // code id in [0, L*E*K)
  int lane = threadIdx.x & 31;
  const float* row = cb + (size_t)wid * DD;
  float s = 0.f;
#pragma unroll
  for (int t = 0; t < 8; t++) { float v = row[t * 32 + lane]; s += v * v; }
  for (int m = 16; m >= 1; m >>= 1) s += __shfl_xor(s, m, 32);
  if (lane == 0) c2[wid] = s;
}

// ---------------------------------------------------------------- gate + residual init
__global__ void k_gate(const float* __restrict__ x, const float* __restrict__ W,
                       const float* __restrict__ b, float* __restrict__ residual,
                       int* __restrict__ expert_id, int* __restrict__ meta,
                       float* __restrict__ out_idx) {
  int n = blockIdx.x * 8 + (threadIdx.x >> 5);
  int lane = threadIdx.x & 31;
  const float* xr = x + (size_t)n * DD;
  float acc[EE];
#pragma unroll
  for (int e = 0; e < EE; e++) acc[e] = 0.f;
#pragma unroll
  for (int t = 0; t < 8; t++) {
    int d = t * 32 + lane;
    float xv = xr[d];
    residual[(size_t)n * DD + d] = xv;
    const float* wr = W + d * EE;
#pragma unroll
    for (int e = 0; e < EE; e++) acc[e] += xv * wr[e];
  }
#pragma unroll
  for (int e = 0; e < EE; e++)
    for (int m = 16; m >= 1; m >>= 1) acc[e] += __shfl_xor(acc[e], m, 32);
  if (lane == 0) {
    int best = 0;
    float bv = acc[0] + b[0];
#pragma unroll
    for (int e = 1; e < EE; e++) {
      float v = acc[e] + b[e];
      if (v > bv) { bv = v; best = e; }     // strict > : first index on ties
    }
    expert_id[n] = best;
    atomicAdd(&meta[best], 1);              // counts
    out_idx[n * (LL + 1) + LL] = (float)best;
  }
}

// ---------------------------------------------------------------- padded prefix sums
// meta[0..7]=counts  meta[8..15]=cursor  meta[16..24]=padded offsets  meta[25]=numTiles
__global__ void k_offsets(int* meta) {
  int po = 0;
#pragma unroll
  for (int e = 0; e < EE; e++) {
    meta[16 + e] = po;
    meta[8 + e]  = po;
    po += ((meta[e] + 15) >> 4) << 4;
  }
  meta[16 + EE] = po;
  meta[25] = po >> 4;
}

// ---------------------------------------------------------------- scatter rows by expert
__global__ void k_scatter(const int* __restrict__ expert_id, int* __restrict__ meta,
                          int* __restrict__ perm) {
  int n = blockIdx.x * blockDim.x + threadIdx.x;
  if (n >= NN) return;
  int e = expert_id[n];
  int pos = atomicAdd(&meta[8 + e], 1);
  perm[pos] = n;
}

// ---------------------------------------------------------------- reset per-row argmin keys
__global__ void k_minreset(unsigned long long* __restrict__ minkey) {
  int n = blockIdx.x * blockDim.x + threadIdx.x;
  if (n < NN) minkey[n] = 0xffffffffffffffffull;
}

// ---------------------------------------------------------------- WMMA distance + argmin
// grid: (MAX_TILES, KK/128); block: 256 threads = 8 waves; wave handles 16 codes.
__global__ void k_dots(const float* __restrict__ cbL, const float* __restrict__ c2L,
                       const float* __restrict__ residual,
                       const int* __restrict__ perm, const int* __restrict__ meta,
                       unsigned long long* __restrict__ minkey) {
  __shared__ float lds[16 * 260];           // stride 260 -> conflict-free A reads
  int numTiles = meta[25];
  int tile = blockIdx.x;
  if (tile >= numTiles) return;             // block-uniform
  int base16 = tile * 16;

  // which expert owns this tile (last e with padded_off[e] <= base16)
  int e = 0;
#pragma unroll
  for (int j = 0; j < EE; j++)
    if (meta[16 + j] <= base16) e = j;

  // stage 16 gathered residual rows into LDS (zeros for padding rows)
  {
    int tid = threadIdx.x;
    int row = tid >> 4;            // 0..15
    int seg = (tid & 15) * 16;     // 0..240
    int n = perm[base16 + row];
#pragma unroll
    for (int j = 0; j < 16; j += 4) {
      float4 v = make_float4(0.f, 0.f, 0.f, 0.f);
      if (n >= 0) v = *(const float4*)(residual + (size_t)n * DD + seg + j);
      lds[row * 260 + seg + j + 0] = v.x;
      lds[row * 260 + seg + j + 1] = v.y;
      lds[row * 260 + seg + j + 2] = v.z;
      lds[row * 260 + seg + j + 3] = v.w;
    }
  }
  __syncthreads();

  int lane = threadIdx.x & 31;
  int wave = threadIdx.x >> 5;
  int code = blockIdx.y * 128 + wave * 16 + (lane & 15);
  int koff = (lane >= 16) ? 2 : 0;          // K pair selector per WMMA f32 layout
  int arow = lane & 15;
  const float* brow = cbL + ((size_t)e * KK + code) * DD;

  v8f acc = {0.f, 0.f, 0.f, 0.f, 0.f, 0.f, 0.f, 0.f};
#pragma unroll 8
  for (int d0 = 0; d0 < DD; d0 += 4) {
    v2f a, b;
    a[0] = lds[arow * 260 + d0 + koff];
    a[1] = lds[arow * 260 + d0 + koff + 1];
    float2 bv = *(const float2*)(brow + d0 + koff);
    b[0] = bv.x;
    b[1] = bv.y;
    // D = A(16x4) * B(4x16) + C, exact fp32
    acc = __builtin_amdgcn_wmma_f32_16x16x4_f32(false, a, false, b, (short)0,
                                                acc, false, false);
  }

  float c2v = c2L[(size_t)e * KK + code];
#pragma unroll
  for (int i = 0; i < 8; i++) {
    float dist = c2v - 2.0f * acc[i];
    unsigned long long key =
        (((unsigned long long)fkey32(dist)) << 32) | (unsigned int)code;
    for (int m = 8; m >= 1; m >>= 1) {
      unsigned long long o = shfl_xor_u64(key, m);
      if (o < key) key = o;
    }
    if ((lane & 15) == 0) {
      int row_local = i + ((lane >> 4) << 3);   // VGPR i: rows i (lanes 0-15), i+8 (16-31)
      int n = perm[base16 + row_local];
      if (n >= 0) atomicMin(&minkey[n], key);
    }
  }
}

// ---------------------------------------------------------------- per-level finalize
__global__ void k_finalize(const float* __restrict__ cbL, float* __restrict__ residual,
                           const int* __restrict__ expert_id,
                           const unsigned long long* __restrict__ minkey,
                           float* __restrict__ xq, float* __restrict__ out_idx,
                           float* __restrict__ loss_rows, int l) {
  int n = blockIdx.x * 8 + (threadIdx.x >> 5);
  int lane = threadIdx.x & 31;
  unsigned long long key = minkey[n];
  int idx = (int)(unsigned int)(key & 0xffffffffull);
  int e = expert_id[n];
  const float* q = cbL + ((size_t)e * KK + idx) * DD;
  float s = 0.f;
#pragma unroll
  for (int t = 0; t < 8; t++) {
    int d = t * 32 + lane;
    float rv = residual[(size_t)n * DD + d];
    float qv = q[d];
    float diff = qv - rv;
    s += diff * diff;
    residual[(size_t)n * DD + d] = rv - qv;   // straight-through: r -= quant
    xq[(size_t)n * DD + d] += qv;             // x_q += quant
  }
  for (int m = 16; m >= 1; m >>= 1) s += __shfl_xor(s, m, 32);
  if (lane == 0) {
    loss_rows[l * NN + n] = s;
    out_idx[n * (LL + 1) + l] = (float)idx;
  }
}

// ---------------------------------------------------------------- deterministic loss reduce
__global__ void k_loss(const float* __restrict__ loss_rows, float* __restrict__ out_loss) {
  __shared__ float sh[256];
  int tid = threadIdx.x;
  float s = 0.f;
  for (int i = tid; i < LL * NN; i += 256) s += loss_rows[i];
  sh[tid] = s;
  __syncthreads();
  for (int w = 128; w >= 1; w >>= 1) {
    if (tid < w) sh[tid] += sh[tid + w];
    __syncthreads();
  }
  // loss_l = (1+BETA)*mean = 2*S_l/(N*D); mean over L levels
  if (tid == 0) out_loss[0] = sh[0] * (2.0f / ((float)LL * (float)NN * (float)DD));
}

// ================================================================ launch
extern "C" void kernel_launch(void* const* d_in, const int* in_sizes, int n_in,
                              void* d_out, int out_size, void* d_ws, size_t ws_size,
                              hipStream_t stream) {
  const float* x  = (const float*)d_in[0];
  // d_in[1] = labels (unused by the reference computation)
  const float* cb = (const float*)d_in[2];   // [L,E,K,D]
  const float* W  = (const float*)d_in[3];   // [D,E]
  const float* bb = (const float*)d_in[4];   // [E]

  float* out      = (float*)d_out;
  float* xq       = out;                     // [N*D]
  float* out_loss = out + (size_t)NN * DD;   // [1]
  float* out_idx  = out_loss + 1;            // [N*(L+1)]

  char* p = (char*)d_ws;
  float* residual = (float*)p;                 p += (size_t)NN * DD * 4;        // 8 MB
  float* c2       = (float*)p;                 p += (size_t)LL * EE * KK * 4;   // 128 KB
  float* loss_rows= (float*)p;                 p += (size_t)LL * NN * 4;        // 128 KB
  unsigned long long* minkey = (unsigned long long*)p; p += (size_t)NN * 8;     // 64 KB
  int* expert_id  = (int*)p;                   p += (size_t)NN * 4;             // 32 KB
  int* perm       = (int*)p;                   p += (size_t)(NN + 256) * 4;
  int* meta       = (int*)p;                   // 32 ints

  k_init<<<(NN * DD + 255) / 256, 256, 0, stream>>>(xq, perm, meta);
  k_c2<<<(LL * EE * KK) / 8, 256, 0, stream>>>(cb, c2);
  k_gate<<<NN / 8, 256, 0, stream>>>(x, W, bb, residual, expert_id, meta, out_idx);
  k_offsets<<<1, 1, 0, stream>>>(meta);
  k_scatter<<<NN / 256, 256, 0, stream>>>(expert_id, meta, perm);

  for (int l = 0; l < LL; l++) {
    const float* cbL = cb + (size_t)l * EE * KK * DD;
    const float* c2L = c2 + (size_t)l * EE * KK;
    k_minreset<<<NN / 256, 256, 0, stream>>>(minkey);
    dim3 grid(MAX_TILES, KK / 128);
    k_dots<<<grid, 256, 0, stream>>>(cbL, c2L, residual, perm, meta, minkey);
    k_finalize<<<NN / 8, 256, 0, stream>>>(cbL, residual, expert_id, minkey,
                                           xq, out_idx, loss_rows, l);
  }
  k_loss<<<1, 256, 0, stream>>>(loss_rows, out_loss);
}